// HGCN_GU_19146964205954
// MI455X (gfx1250) — compile-verified
//
#include <hip/hip_runtime.h>

#define NGROUPS 50000
#define NUSERS  200000
#define DIM     128
#define LAYERS  3

typedef __attribute__((ext_vector_type(2))) float v2f;
typedef __attribute__((ext_vector_type(8))) float v8f;

// One wave32 per edge. Lane l owns dims [4l, 4l+4).
// gather_by_cols=1: dst[rows[e]] += vals[e]*src[cols[e]]   (H @ x)
// gather_by_cols=0: dst[cols[e]] += vals[e]*src[rows[e]]   (H^T @ x)
__global__ void spmm_scatter_kernel(const int* __restrict__ rows,
                                    const int* __restrict__ cols,
                                    const float* __restrict__ vals,
                                    const float* __restrict__ src,
                                    float* __restrict__ dst,
                                    int nnz, int gather_by_cols) {
  int e    = (int)((blockIdx.x * (unsigned)blockDim.x + threadIdx.x) >> 5);
  int lane = threadIdx.x & 31;
  if (e >= nnz) return;
  int r = rows[e], c = cols[e];
  float v = vals[e];
  int gi = gather_by_cols ? c : r;
  int si = gather_by_cols ? r : c;
  const float4* sp = (const float4*)(src + (long)gi * DIM);
  float4 x = sp[lane];
  float* dp = dst + (long)si * DIM + lane * 4;
  atomicAdd(dp + 0, v * x.x);
  atomicAdd(dp + 1, v * x.y);
  atomicAdd(dp + 2, v * x.z);
  atomicAdd(dp + 3, v * x.w);
}

// msg = [node_msg, node_msg*group] @ W + b  via V_WMMA_F32_16X16X4_F32.
// Block = 256 threads = 8 waves; block owns 16 M-rows, wave w owns N-tile w.
// K-loop split at the concat boundary so both halves are branch-free.
__global__ void gemm_wmma_kernel(const float* __restrict__ node_msg, // [M,128]
                                 const float* __restrict__ group,    // [M,128]
                                 const float* __restrict__ W,        // [256,128]
                                 const float* __restrict__ bias,     // [128]
                                 float* __restrict__ out) {          // [M,128]
  const int m0   = blockIdx.x * 16;
  const int wave = threadIdx.x >> 5;   // 0..7 -> N tile
  const int lane = threadIdx.x & 31;
  const int half = lane >> 4;          // selects K pair within the 4-wide step
  const int l15  = lane & 15;
  const int n0   = wave * 16;

  // A/B fragment layout for V_WMMA_F32_16X16X4_F32:
  //   lanes 0-15  hold K = k0+0, k0+1   (VGPR0, VGPR1)
  //   lanes 16-31 hold K = k0+2, k0+3
  const float2* arow2 = (const float2*)(node_msg + (long)(m0 + l15) * DIM);
  const float2* grow2 = (const float2*)(group    + (long)(m0 + l15) * DIM);
  const float*  bcol  = W + n0 + l15;

  v8f acc = {};

  // K = 0..127: A = node_msg
#pragma unroll 8
  for (int k0 = 0; k0 < DIM; k0 += 4) {
    const int kb = k0 + 2 * half;
    const float2 av = arow2[kb >> 1];
    v2f a, b;
    a.x = av.x;
    a.y = av.y;
    b.x = bcol[(long)kb * DIM];
    b.y = bcol[(long)(kb + 1) * DIM];
    acc = __builtin_amdgcn_wmma_f32_16x16x4_f32(false, a, false, b,
                                                (short)0, acc, false, false);
  }

  // K = 128..255: A = node_msg * group (gated half of the concat)
#pragma unroll 8
  for (int k0 = 0; k0 < DIM; k0 += 4) {
    const int kk = k0 + 2 * half;        // index within the 128-wide row
    const int kb = DIM + kk;             // index within W's 256-wide K
    const float2 av = arow2[kk >> 1];
    const float2 gv = grow2[kk >> 1];
    v2f a, b;
    a.x = av.x * gv.x;
    a.y = av.y * gv.y;
    b.x = bcol[(long)kb * DIM];
    b.y = bcol[(long)(kb + 1) * DIM];
    acc = __builtin_amdgcn_wmma_f32_16x16x4_f32(false, a, false, b,
                                                (short)0, acc, false, false);
  }

  const float bv = bias[n0 + l15];
#pragma unroll
  for (int r = 0; r < 8; ++r) {
    const int m = m0 + r + half * 8;   // C/D layout: VGPR r -> M=r (lanes 0-15), M=r+8 (16-31)
    out[(long)m * DIM + n0 + l15] = acc[r] + bv;
  }
}

__global__ void add4_kernel(float4* __restrict__ dst, const float4* __restrict__ src, int n4) {
  int i = (int)(blockIdx.x * (unsigned)blockDim.x + threadIdx.x);
  if (i >= n4) return;
  float4 d = dst[i], s = src[i];
  d.x += s.x; d.y += s.y; d.z += s.z; d.w += s.w;
  dst[i] = d;
}

extern "C" void kernel_launch(void* const* d_in, const int* in_sizes, int n_in,
                              void* d_out, int out_size, void* d_ws, size_t ws_size,
                              hipStream_t stream) {
  const float* group_in = (const float*)d_in[0];
  const float* user_in  = (const float*)d_in[1];
  const int*   rows     = (const int*)d_in[2];
  const int*   cols     = (const int*)d_in[3];
  const float* vals     = (const float*)d_in[4];
  const float* Ws       = (const float*)d_in[5];
  const float* bs       = (const float*)d_in[6];
  const int nnz = in_sizes[2];

  float* node_sum = (float*)d_out;                        // [NUSERS, DIM]
  float* edge_sum = node_sum + (size_t)NUSERS * DIM;      // [NGROUPS, DIM]

  const size_t gbytes = (size_t)NGROUPS * DIM * sizeof(float); // 25.6 MB
  const size_t ubytes = (size_t)NUSERS  * DIM * sizeof(float); // 102.4 MB
  char* ws = (char*)d_ws;
  float* node_msg = (float*)(ws);
  float* msgA     = (float*)(ws + gbytes);
  float* msgB     = (float*)(ws + 2 * gbytes);
  float* U        = (float*)(ws + 3 * gbytes);            // single user scratch

  // Running sums start at the inputs.
  hipMemcpyAsync(node_sum, user_in,  ubytes, hipMemcpyDeviceToDevice, stream);
  hipMemcpyAsync(edge_sum, group_in, gbytes, hipMemcpyDeviceToDevice, stream);

  dim3 spmmGrid((nnz + 7) / 8);       // 8 waves (edges) per 256-thread block
  const int un4 = NUSERS * DIM / 4;
  const int gn4 = NGROUPS * DIM / 4;

  const float* user_cur  = user_in;
  const float* group_cur = group_in;
  for (int l = 0; l < LAYERS; ++l) {
    float* msg_out = (l & 1) ? msgB : msgA;

    // node_msg = H @ user_cur
    hipMemsetAsync(node_msg, 0, gbytes, stream);
    spmm_scatter_kernel<<<spmmGrid, 256, 0, stream>>>(rows, cols, vals,
                                                      user_cur, node_msg, nnz, 1);
    // msg = [node_msg, node_msg*group_cur] @ W_l + b_l
    gemm_wmma_kernel<<<NGROUPS / 16, 256, 0, stream>>>(node_msg, group_cur,
                                                       Ws + (size_t)l * 2 * DIM * DIM,
                                                       bs + (size_t)l * DIM, msg_out);
    // user_next = H^T @ msg  (user_cur already fully consumed; reuse U safely)
    hipMemsetAsync(U, 0, ubytes, stream);
    spmm_scatter_kernel<<<spmmGrid, 256, 0, stream>>>(rows, cols, vals,
                                                      msg_out, U, nnz, 0);
    // running sums
    add4_kernel<<<(un4 + 255) / 256, 256, 0, stream>>>((float4*)node_sum,
                                                       (const float4*)U, un4);
    add4_kernel<<<(gn4 + 255) / 256, 256, 0, stream>>>((float4*)edge_sum,
                                                       (const float4*)msg_out, gn4);
    user_cur  = U;
    group_cur = msg_out;
  }
}